// RCNNNet_39797166965281
// MI455X (gfx1250) — compile-verified
//
#include <hip/hip_runtime.h>
#include <hip/hip_bf16.h>
#include <stdint.h>

// ---------------- problem constants (from reference) ----------------
#define OUTR     12
#define NCELL    (OUTR * OUTR * OUTR)   // 1728 cells per ROI
#define NB       2
#define NROI     128
#define NPTS     50000
#define CPART    4
#define CRPN     16

// point-tile staging via TDM: 500 points * 3 f32 = 1500 f32 = 6000 B
#define TILE_PTS 500
#define TILE_F   (TILE_PTS * 3)
#define NTILES   (NPTS / TILE_PTS)      // exactly 100, no remainder
#define THREADS  256                    // 8 wave32s

// LDS layout (floats): cnt[1728] | sum[1728*4] | maxkey[1728*16] | pts[2*1500]
#define SMEM_FLOATS (NCELL * 21 + 2 * TILE_F)
#define SMEM_BYTES  ((size_t)SMEM_FLOATS * 4)

typedef unsigned int v4u __attribute__((ext_vector_type(4)));
typedef int          v8i __attribute__((ext_vector_type(8)));
typedef int          v4i __attribute__((ext_vector_type(4)));

// Monotonic order-preserving key for f32 max via unsigned LDS atomics.
// All real floats map to keys > 0, so key 0 is the "empty" identity.
__device__ __forceinline__ uint32_t fkey(float f) {
    uint32_t u = __float_as_uint(f);
    return (u & 0x80000000u) ? ~u : (u | 0x80000000u);
}
__device__ __forceinline__ float fdec(uint32_t k) {
    uint32_t u = (k & 0x80000000u) ? (k & 0x7FFFFFFFu) : ~k;
    return __uint_as_float(u);
}

// One TDM descriptor = 1-D tile of TILE_F f32 starting at gsrc, copied to LDS at byte address lds_addr.
// D# packing per CDNA5 ISA ch.8 (group0: count/lds/global/type, group1: dims).
// This toolchain's builtin takes 6 operands:
//   (uint32x4 g0, int32x8 g1, int32x4 g2, int32x4 g3, int32x8 extra, i32 cpol)
// g2/g3/extra are only meaningful for >2D tensors / gather mode -> zero here.
__device__ __forceinline__ void tdm_load_tile(uint32_t lds_addr, const float* gsrc) {
    uint64_t ga = (uint64_t)(uintptr_t)gsrc;
    v4u g0;
    g0[0] = 1u;                                              // count=1, user mode
    g0[1] = lds_addr;                                        // LDS byte address
    g0[2] = (uint32_t)ga;                                    // global_addr[31:0]
    g0[3] = (uint32_t)((ga >> 32) & 0x01FFFFFFu) | (2u << 30); // addr[56:32] | type=2
    v8i g1;
    const uint32_t td0 = (uint32_t)(NPTS * 3);               // tensor_dim0 = 150000 f32
    g1[0] = (int)(2u << 16);                                 // data_size = 4 bytes
    g1[1] = (int)((td0 & 0xFFFFu) << 16);                    // tensor_dim0[15:0]
    g1[2] = (int)(((td0 >> 16) & 0xFFFFu) | (1u << 16));     // tensor_dim0[31:16], tensor_dim1=1
    g1[3] = (int)((uint32_t)TILE_F << 16);                   // tile_dim0 = 1500 (4B-multiple)
    g1[4] = 0;                                               // tile_dim1=0, tile_dim2=0 (unused)
    g1[5] = (int)td0;                                        // tensor_dim0_stride[31:0]
    g1[6] = 0;                                               // stride[47:32], dim1_stride lo
    g1[7] = 0;
    v4i z4  = {0, 0, 0, 0};                                  // groups 2/3 unused (<=2D)
    v8i z8  = {0, 0, 0, 0, 0, 0, 0, 0};
    __builtin_amdgcn_tensor_load_to_lds(g0, g1, z4, z4, z8, 0);
}

__global__ void __launch_bounds__(THREADS)
rcnn_roi_pool_kernel(const float* __restrict__ rois,         // (B,N,7)
                     const float* __restrict__ pts,          // (B,M,3)
                     const float* __restrict__ pfeat,        // (B,M,4)
                     const float* __restrict__ rfeat,        // (B,M,16)
                     float* __restrict__ out)                // part | rpn, concat
{
    extern __shared__ float smem[];
    uint32_t* s_cnt = (uint32_t*)smem;                       // [1728]
    float*    s_sum = smem + NCELL;                          // [1728*4]
    uint32_t* s_max = (uint32_t*)(smem + NCELL * 5);         // [1728*16]
    float*    s_pts = smem + NCELL * 21;                     // [2*1500]

    const int g   = blockIdx.x;                              // global ROI id, 0..255
    const int b   = g >> 7;                                  // batch = g / 128
    const int tid = threadIdx.x;

    const float*  bpts = pts + (size_t)b * NPTS * 3;
    const float4* bpf  = (const float4*)(pfeat + (size_t)b * NPTS * CPART);
    const float4* brf  = (const float4*)(rfeat + (size_t)b * NPTS * CRPN);

    // Low 32 bits of the generic LDS pointer == wave-relative LDS byte offset.
    const uint32_t lds_pts = (uint32_t)(uintptr_t)(void*)s_pts;

    // Prologue: wave 0 kicks off tile 0 DMA before we touch accumulators.
    if (tid < 32) tdm_load_tile(lds_pts, bpts);

    // Zero accumulators (key 0 == -inf sentinel for max).
    for (int i = tid; i < NCELL;      i += THREADS) s_cnt[i] = 0u;
    for (int i = tid; i < NCELL * 4;  i += THREADS) s_sum[i] = 0.0f;
    for (int i = tid; i < NCELL * 16; i += THREADS) s_max[i] = 0u;

    // ROI params (uniform across the workgroup -> scalar loads).
    const float* R = rois + (size_t)g * 7;
    const float cx = R[0], cy = R[1], cz = R[2];
    const float dx = R[3], dy = R[4], dz = R[5];
    const float rz = R[6];
    const float cr = cosf(rz), sr = sinf(rz);
    const float hx = 0.5f * dx, hy = 0.5f * dy, hz = 0.5f * dz;
    const float ivx = (float)OUTR / dx, ivy = (float)OUTR / dy, ivz = (float)OUTR / dz;

    __syncthreads();

    // Double-buffered TDM pipeline over point tiles.
    for (int t = 0; t < NTILES; ++t) {
        if (tid < 32) {
            if (t + 1 < NTILES) {
                tdm_load_tile(lds_pts + (uint32_t)((t + 1) & 1) * (TILE_F * 4u),
                              bpts + (size_t)(t + 1) * TILE_F);
                __builtin_amdgcn_s_wait_tensorcnt(1);        // tile t landed
            } else {
                __builtin_amdgcn_s_wait_tensorcnt(0);
            }
        }
        __syncthreads();                                     // publish tile t to all waves

        const float* tp = s_pts + (t & 1) * TILE_F;
        for (int j = tid; j < TILE_PTS; j += THREADS) {
            const float x = tp[j * 3 + 0];
            const float y = tp[j * 3 + 1];
            const float z = tp[j * 3 + 2];
            const float sx = x - cx, sy = y - cy, szl = z - cz;
            const float lx = sx * cr + sy * sr;
            const float ly = sy * cr - sx * sr;
            if (fabsf(lx) < hx && fabsf(ly) < hy && fabsf(szl) < hz) {
                int vx = (int)floorf((lx  + hx) * ivx);
                int vy = (int)floorf((ly  + hy) * ivy);
                int vz = (int)floorf((szl + hz) * ivz);
                vx = min(OUTR - 1, max(0, vx));
                vy = min(OUTR - 1, max(0, vy));
                vz = min(OUTR - 1, max(0, vz));
                const int cell = (vx * OUTR + vy) * OUTR + vz;
                const int pt   = t * TILE_PTS + j;

                atomicAdd(&s_cnt[cell], 1u);                 // ds_add_u32
                const float4 p4 = bpf[pt];
                atomicAdd(&s_sum[cell * 4 + 0], p4.x);       // ds_add_f32 x4
                atomicAdd(&s_sum[cell * 4 + 1], p4.y);
                atomicAdd(&s_sum[cell * 4 + 2], p4.z);
                atomicAdd(&s_sum[cell * 4 + 3], p4.w);
#pragma unroll
                for (int q = 0; q < 4; ++q) {
                    const float4 r4 = brf[pt * 4 + q];
                    atomicMax(&s_max[cell * 16 + q * 4 + 0], fkey(r4.x)); // ds_max_u32
                    atomicMax(&s_max[cell * 16 + q * 4 + 1], fkey(r4.y));
                    atomicMax(&s_max[cell * 16 + q * 4 + 2], fkey(r4.z));
                    atomicMax(&s_max[cell * 16 + q * 4 + 3], fkey(r4.w));
                }
            }
        }
        __syncthreads();                                     // retire buffer before reuse
    }

    // Finalize: avg-pool (sum/max(cnt,1)) and max-pool (0 where empty), vector stores.
    float4* out_part = (float4*)out + (size_t)g * NCELL;                       // 1 float4/cell
    float4* out_rpn  = (float4*)(out + (size_t)NB * NROI * NCELL * CPART)
                     + (size_t)g * NCELL * 4;                                  // 4 float4/cell
    for (int cell = tid; cell < NCELL; cell += THREADS) {
        const uint32_t c  = s_cnt[cell];
        const float    iv = 1.0f / fmaxf((float)c, 1.0f);
        float4 s4;
        s4.x = s_sum[cell * 4 + 0] * iv;
        s4.y = s_sum[cell * 4 + 1] * iv;
        s4.z = s_sum[cell * 4 + 2] * iv;
        s4.w = s_sum[cell * 4 + 3] * iv;
        out_part[cell] = s4;
        const bool nz = (c > 0u);
#pragma unroll
        for (int q = 0; q < 4; ++q) {
            float4 m4;
            m4.x = nz ? fdec(s_max[cell * 16 + q * 4 + 0]) : 0.0f;
            m4.y = nz ? fdec(s_max[cell * 16 + q * 4 + 1]) : 0.0f;
            m4.z = nz ? fdec(s_max[cell * 16 + q * 4 + 2]) : 0.0f;
            m4.w = nz ? fdec(s_max[cell * 16 + q * 4 + 3]) : 0.0f;
            out_rpn[cell * 4 + q] = m4;
        }
    }
}

extern "C" void kernel_launch(void* const* d_in, const int* in_sizes, int n_in,
                              void* d_out, int out_size, void* d_ws, size_t ws_size,
                              hipStream_t stream) {
    (void)in_sizes; (void)n_in; (void)out_size; (void)d_ws; (void)ws_size;
    const float* rois  = (const float*)d_in[0];   // (2,128,7)
    const float* vcen  = (const float*)d_in[1];   // (2,50000,3)
    const float* pfeat = (const float*)d_in[2];   // (2,50000,4)
    const float* rfeat = (const float*)d_in[3];   // (2,50000,16)
    float* out = (float*)d_out;

    // 157 KB LDS per workgroup (CDNA5 WGP has 320 KB) — raise the opt-in cap.
    (void)hipFuncSetAttribute(reinterpret_cast<const void*>(rcnn_roi_pool_kernel),
                              hipFuncAttributeMaxDynamicSharedMemorySize, (int)SMEM_BYTES);

    rcnn_roi_pool_kernel<<<NB * NROI, THREADS, SMEM_BYTES, stream>>>(
        rois, vcen, pfeat, rfeat, out);
}